// PriorW_12103217840186
// MI455X (gfx1250) — compile-verified
//
#include <hip/hip_runtime.h>

// PriorW: batched 16x16 triangular-inverse + row-normalize, then
// W[b] (4096x16) = Z[b] (4096x16) @ M[b]^T (16x16), f32 throughout.
// Memory-bound (AI = 4 FLOP/B): ~537 MB mandatory traffic -> ~23 us at
// 23.3 TB/s. V_WMMA_F32_16X16X4_F32 is the matrix path with zero precision
// loss; compute needs only ~93 TF f32 to keep pace with HBM.

typedef __attribute__((ext_vector_type(2))) float v2f;
typedef __attribute__((ext_vector_type(8))) float v8f;

#define WDIM   16
#define NBATCH 1024
#define NROWS  4096

// ---------------------------------------------------------------------------
// Kernel 1: per-batch Bartlett factor -> B = M^T (stored row-major [j][i]).
// One wave per batch. Lanes 0..15 each solve one column of inv(A) by forward
// substitution (fully unrolled so x[] stays in VGPRs), then normalize.
// ---------------------------------------------------------------------------
__global__ __launch_bounds__(32) void prep_M_kernel(
    const float* __restrict__ A_noise,   // [NBATCH,16,16]
    const float* __restrict__ chisq,     // [NBATCH,16]
    float* __restrict__ Bout)            // [NBATCH,16,16] = M^T
{
    const int b    = blockIdx.x;
    const int lane = threadIdx.x;        // 0..31 (wave32)

    __shared__ float A[WDIM][WDIM];

    // Build A = tril(A_noise,-1) + diag(sqrt(chisq)) in LDS.
    #pragma unroll
    for (int t = 0; t < 8; ++t) {
        int e = lane + 32 * t;           // 0..255
        int i = e >> 4, j = e & 15;
        float v;
        if (j < i)       v = A_noise[(size_t)b * 256 + e];
        else if (j == i) v = sqrtf(chisq[(size_t)b * WDIM + i]);
        else             v = 0.0f;
        A[i][j] = v;
    }
    __syncthreads();

    if (lane < WDIM) {
        const int c = lane;              // column of inv(A) == row of At/M
        float x[WDIM];
        // Forward substitution, fully unrolled: static indices keep x[] in
        // registers. For k < c the recurrence yields exact zeros.
        #pragma unroll
        for (int k = 0; k < WDIM; ++k) {
            float s = (k == c) ? 1.0f : 0.0f;
            #pragma unroll
            for (int j = 0; j < WDIM; ++j) {
                if (j < k) s -= A[k][j] * x[j];
            }
            x[k] = s / A[k][k];
        }
        float nrm = 0.0f;
        #pragma unroll
        for (int k = 0; k < WDIM; ++k) nrm += x[k] * x[k];
        const float sc = rsqrtf(nrm);
        // M[c][j] = sc * x[j];  store B = M^T:  B[j][c] = M[c][j].
        float* Bb = Bout + (size_t)b * 256;
        #pragma unroll
        for (int j = 0; j < WDIM; ++j) Bb[j * WDIM + c] = sc * x[j];
    }
}

// ---------------------------------------------------------------------------
// Kernel 2: W = Z @ B using V_WMMA_F32_16X16X4_F32 (four chained K=4 steps).
// 1 block per batch, 8 waves/block; each wave sweeps 32 of the 256 16-row
// tiles. B (16x16) is loaded once per wave into the WMMA B layout. The wave
// id is scalarized via readfirstlane so the tile loop is a uniform SGPR loop
// (no exec-mask loop machinery, EXEC stays all-1s for WMMA), and unroll(2)
// lets the compiler pipeline two tiles' loads/WMMAs/stores.
//
// Layouts (CDNA5 ISA, 32-bit operands, wave32):
//   A 16x4:  lanes 0-15 = rows, VGPR0/1 = K0/K1; lanes 16-31 = K2/K3.
//   B 4x16:  lanes = N columns; VGPR0/1 = K rows, wave-half split as A.
//   C/D 16x16: VGPR v -> M=v (lanes 0-15), M=8+v (lanes 16-31); lane&15 = N.
// ---------------------------------------------------------------------------
__global__ __launch_bounds__(256) void gemm_wmma_kernel(
    const float* __restrict__ z,     // [NBATCH, NROWS, 16]
    const float* __restrict__ Bmat,  // [NBATCH, 16, 16] = M^T
    float* __restrict__ w)           // [NBATCH, NROWS, 16]
{
    const int b    = blockIdx.x;
    const int lane = threadIdx.x & 31;
    // Scalar (SGPR) wave id -> uniform tile loop.
    const int wave = __builtin_amdgcn_readfirstlane(threadIdx.x >> 5); // 0..7
    const int n    = lane & 15;           // column within tile
    const int hi   = lane >> 4;           // wave half: 0 or 1

    const float* zb = z + (size_t)b * NROWS * WDIM;
    float*       wb = w + (size_t)b * NROWS * WDIM;
    const float* Bb = Bmat + (size_t)b * 256;

    // B operand: four K=4 chunks, 2 VGPRs each (loaded once per wave).
    v2f bReg[4];
    #pragma unroll
    for (int kc = 0; kc < 4; ++kc) {
        const int r = kc * 4 + 2 * hi;    // K row for VGPR0 of this chunk
        v2f t;
        t.x = Bb[r * WDIM + n];
        t.y = Bb[(r + 1) * WDIM + n];
        bReg[kc] = t;
    }

    // 256 tiles per batch, strided across the 8 waves -> 32 iterations each.
    #pragma unroll 2
    for (int t = wave; t < 256; t += 8) {
        const int    row0  = t * WDIM;
        const float* ztile = zb + (size_t)row0 * WDIM;

        // A operand: per-lane float2 loads matching the 16x4 layout.
        v2f aReg[4];
        #pragma unroll
        for (int kc = 0; kc < 4; ++kc) {
            aReg[kc] = *(const v2f*)(ztile + (size_t)n * WDIM + kc * 4 + 2 * hi);
        }

        // Unconditional speculative prefetch of this wave's next tile
        // (overshoot past the final batch is silently dropped per ISA).
        __builtin_prefetch(ztile + 8 * WDIM * WDIM + (size_t)n * WDIM, 0, 1);

        // Chain four 16x16x4 f32 WMMAs over K = 16.
        v8f c = {0.f, 0.f, 0.f, 0.f, 0.f, 0.f, 0.f, 0.f};
        #pragma unroll
        for (int kc = 0; kc < 4; ++kc) {
            c = __builtin_amdgcn_wmma_f32_16x16x4_f32(
                    /*neg_a=*/false, aReg[kc],
                    /*neg_b=*/false, bReg[kc],
                    /*c_mod=*/(short)0, c,
                    /*reuse_a=*/false, /*reuse_b=*/false);
        }

        // Scatter D per the C/D VGPR layout.
        float* wtile = wb + (size_t)row0 * WDIM;
        #pragma unroll
        for (int v = 0; v < 8; ++v) {
            wtile[(size_t)(hi * 8 + v) * WDIM + n] = c[v];
        }
    }
}

// ---------------------------------------------------------------------------
// Host-side launcher. d_ws needs NBATCH*256*4 = 1 MB for the B matrices.
// ---------------------------------------------------------------------------
extern "C" void kernel_launch(void* const* d_in, const int* in_sizes, int n_in,
                              void* d_out, int out_size, void* d_ws, size_t ws_size,
                              hipStream_t stream) {
    const float* A_noise = (const float*)d_in[0];  // [1024,16,16]
    const float* chisq   = (const float*)d_in[1];  // [1024,16]
    const float* z       = (const float*)d_in[2];  // [1024,4096,16]
    float*       wout    = (float*)d_out;          // [1024,4096,16]
    float*       Bmat    = (float*)d_ws;           // [1024,16,16] scratch

    prep_M_kernel<<<NBATCH, 32, 0, stream>>>(A_noise, chisq, Bmat);
    gemm_wmma_kernel<<<NBATCH, 256, 0, stream>>>(z, Bmat, wout);
}